// PS3AttentionLayer_9242769621431
// MI455X (gfx1250) — compile-verified
//
#include <hip/hip_runtime.h>
#include <hip/hip_bf16.h>

typedef float v2f __attribute__((ext_vector_type(2)));
typedef float v8f __attribute__((ext_vector_type(8)));

#define NBAT   4
#define NSEQ   2048
#define DIMC   512
#define HEADS  6
#define DHEAD  64
#define NQKV   1152        // 3 * 6 * 64
#define PSEG   281
#define ESEG   297         // P + M
#define QSCALE 0.125f      // 64^-0.5
#define LN_EPS 1e-5f

#define XSTR   516         // LDS stride for xn tile (step mod 64 = 4 -> conflict-free frags)
#define SSTR   2050        // LDS stride for score rows (even -> b64-aligned A-frags)
#define QSTR   68          // LDS stride for Q tile
#define PSTR   65          // partial-output row stride
#define VWIN   48          // conv V window rows: 16 queries + 32 halo

// ---------- wave32 16-wide shuffle reductions (rows own contiguous half-waves) ----
static __device__ __forceinline__ float red_sum16(float v) {
    v += __shfl_xor(v, 1, 16);
    v += __shfl_xor(v, 2, 16);
    v += __shfl_xor(v, 4, 16);
    v += __shfl_xor(v, 8, 16);
    return v;
}
static __device__ __forceinline__ float red_max16(float v) {
    v = fmaxf(v, __shfl_xor(v, 1, 16));
    v = fmaxf(v, __shfl_xor(v, 2, 16));
    v = fmaxf(v, __shfl_xor(v, 4, 16));
    v = fmaxf(v, __shfl_xor(v, 8, 16));
    return v;
}

// ---------- fp32 WMMA: D = A(16x4) * B(4x16) + C ---------------------------------
static __device__ __forceinline__ v8f wmma_f32(v2f a, v2f b, v8f c) {
    return __builtin_amdgcn_wmma_f32_16x16x4_f32(false, a, false, b, (short)0, c,
                                                 false, false);
}

// in-place softmax over row[lo,hi), 16 cooperating threads (sub = 0..15)
static __device__ __forceinline__ void softmax_range(float* row, int lo, int hi, int sub) {
    float m = -INFINITY;
    for (int c = lo + sub; c < hi; c += 16) m = fmaxf(m, row[c]);
    m = red_max16(m);
    float s = 0.f;
    for (int c = lo + sub; c < hi; c += 16) {
        float e = __expf(row[c] - m);
        row[c] = e;
        s += e;
    }
    s = red_sum16(s);
    float inv = 1.0f / s;
    for (int c = lo + sub; c < hi; c += 16) row[c] *= inv;
}

// =================================================================================
// Kernel 1: fused LayerNorm + QKV GEMM.
//   grid = (9, 512): blockIdx.y -> 16-row tile of the 8192 rows,
//                    blockIdx.x -> 128-col tile of the 1152 output cols.
//   8 waves x 16 cols each, K = 512 via WMMA f32 16x16x4.
// =================================================================================
__global__ __launch_bounds__(256)
void ps3_ln_qkv_kernel(const float* __restrict__ x,
                       const float* __restrict__ gamma,
                       const float* __restrict__ beta,
                       const float* __restrict__ wqkv,
                       float* __restrict__ wsq,
                       float* __restrict__ wsk,
                       float* __restrict__ wsv) {
    __shared__ float xn[16 * XSTR];

    const int tid  = threadIdx.x;
    const int lane = tid & 31;
    const int l16  = lane & 15;
    const int lh   = lane >> 4;
    const int wave = tid >> 5;

    const int r0 = blockIdx.y * 16;

    // ---- LayerNorm of 16 rows into LDS (16 threads per row) ----
    {
        const int row = tid >> 4;
        const int sub = tid & 15;
        const float* xr = x + (size_t)(r0 + row) * DIMC;
        float s = 0.f, ss = 0.f;
        for (int c = sub; c < DIMC; c += 16) {
            float v = xr[c];
            s += v; ss += v * v;
        }
        s  = red_sum16(s);
        ss = red_sum16(ss);
        const float mu   = s * (1.0f / DIMC);
        const float var  = ss * (1.0f / DIMC) - mu * mu;
        const float rstd = rsqrtf(var + LN_EPS);
        for (int c = sub; c < DIMC; c += 16) {
            float v = xr[c];
            xn[row * XSTR + c] = (v - mu) * rstd * gamma[c] + beta[c];
        }
    }
    __syncthreads();

    // ---- GEMM: this wave's 16 output columns ----
    const int c0 = blockIdx.x * 128 + wave * 16;
    v8f acc = {0.f, 0.f, 0.f, 0.f, 0.f, 0.f, 0.f, 0.f};
    const float* xrow = &xn[l16 * XSTR];
    for (int k0 = 0; k0 < DIMC; k0 += 4) {
        v2f a = *(const v2f*)(xrow + k0 + 2 * lh);              // ds_load_b64
        const float* wp = wqkv + (size_t)(k0 + 2 * lh) * NQKV + c0 + l16;
        v2f b;
        b.x = wp[0];
        b.y = wp[NQKV];
        acc = wmma_f32(a, b, acc);
    }

    // ---- scatter into q/k/v workspace, layout [b][h][n][64] ----
    const int c    = c0 + l16;
    const int part = c / 384;          // uniform per wave's tile
    const int rem  = c - part * 384;
    const int h    = rem >> 6;
    const int d    = rem & 63;
    float* dst = (part == 0) ? wsq : (part == 1) ? wsk : wsv;
    const float sc = (part == 0) ? QSCALE : 1.0f;
#pragma unroll
    for (int r = 0; r < 8; ++r) {
        const int rowg = r0 + r + 8 * lh;
        const int bb = rowg >> 11;       // / 2048
        const int nn = rowg & 2047;
        dst[((size_t)(bb * HEADS + h) * NSEQ + nn) * DHEAD + d] = acc[r] * sc;
    }
}

// =================================================================================
// Kernel 2: fused attention (segmented double softmax) + depthwise conv + output.
//   grid = (128, 24): blockIdx.x -> 16-query tile, blockIdx.y -> (b*6 + h).
//   LDS ~156 KB -> 2 workgroups (16 waves) per 320 KB WGP.
//   Phase 1: S = Q K^T, full 2048-key rows in LDS (+ global_prefetch of next K tile).
//   Phase 2: PS3 softmax (inner softmax on key sub-ranges for h/t query rows, then
//            outer softmax over the full row) -- shuffle-only, no barriers.
//   Phase 3: O = S V; each wave owns one (K-half, d-tile) pair; 2-way LDS reduce.
//   Phase 4: + 33-tap depthwise conv from the LDS-staged V window, store output.
// =================================================================================
__global__ __launch_bounds__(256)
void ps3_attn_kernel(const float* __restrict__ Qg,
                     const float* __restrict__ Kg,
                     const float* __restrict__ Vg,
                     const float* __restrict__ resw,
                     float* __restrict__ out) {
    extern __shared__ float smem[];
    float* S   = smem;                   // 16 x SSTR scores
    float* q_s = S + 16 * SSTR;          // 16 x QSTR Q tile
    float* Pp  = q_s + 16 * QSTR;        // 2 x 16 x PSTR partial output tiles
    float* v_s = Pp + 2 * 16 * PSTR;     // VWIN x 64 V window for the conv

    const int tid  = threadIdx.x;
    const int lane = tid & 31;
    const int l16  = lane & 15;
    const int lh   = lane >> 4;
    const int wave = tid >> 5;

    const int q0 = blockIdx.x * 16;
    const int bh = blockIdx.y;
    const float* Q = Qg + (size_t)bh * NSEQ * DHEAD;
    const float* K = Kg + (size_t)bh * NSEQ * DHEAD;
    const float* V = Vg + (size_t)bh * NSEQ * DHEAD;

    // ---- stage Q tile (16x64) and conv V window (48x64) into LDS, coalesced ----
    for (int e = tid; e < 16 * DHEAD; e += 256) {
        const int m = e >> 6, k = e & 63;
        q_s[m * QSTR + k] = Q[(size_t)(q0 + m) * DHEAD + k];
    }
    for (int e = tid; e < VWIN * DHEAD; e += 256) {
        const int r = e >> 6, d = e & 63;
        const int src = q0 - 16 + r;
        v_s[e] = (src >= 0 && src < NSEQ) ? V[(size_t)src * DHEAD + d] : 0.f;
    }
    __syncthreads();

    // ---- Phase 1: scores. Each wave: 16 key-tiles of 16 over 16 iterations ----
    for (int it = 0; it < 16; ++it) {
        const int j0 = it * 128 + wave * 16;
        if (it + 1 < 16)   // prefetch next K tile (global_prefetch_b8)
            __builtin_prefetch(K + (size_t)(j0 + 128 + l16) * DHEAD + 16 * lh, 0, 0);
        v8f acc = {0.f, 0.f, 0.f, 0.f, 0.f, 0.f, 0.f, 0.f};
        const float* qrow = &q_s[l16 * QSTR];
        const float* krow = K + (size_t)(j0 + l16) * DHEAD;
        for (int k0 = 0; k0 < DHEAD; k0 += 4) {
            v2f a = *(const v2f*)(qrow + k0 + 2 * lh);          // ds_load_b64
            v2f b = *(const v2f*)(krow + k0 + 2 * lh);          // global_load_b64
            acc = wmma_f32(a, b, acc);
        }
#pragma unroll
        for (int r = 0; r < 8; ++r)
            S[(r + 8 * lh) * SSTR + j0 + l16] = acc[r];
    }
    __syncthreads();

    // ---- Phase 2: PS3 segmented softmax, 16 threads per query row ----
    {
        const int row = tid >> 4;
        const int sub = tid & 15;
        float* r = &S[row * SSTR];
        const int i = q0 + row;
        if (i >= PSEG && i < ESEG) {            // 'h' rows: inner softmax on p & t blocks
            softmax_range(r, 0, PSEG, sub);
            softmax_range(r, ESEG, NSEQ, sub);
        } else if (i >= ESEG) {                 // 't' rows: inner softmax on t block
            softmax_range(r, ESEG, NSEQ, sub);
        }
        softmax_range(r, 0, NSEQ, sub);         // outer softmax over all 2048 keys
    }
    __syncthreads();

    // ---- Phase 3: O = S @ V. wave -> (K-half kh, d-tile t); 1024 keys each ----
    {
        const int kh = wave >> 2;                // 0/1: keys [kh*1024, kh*1024+1024)
        const int t  = wave & 3;                 // d-tile: d0 = t*16
        const int d0 = t * 16;
        v8f acc = {0.f, 0.f, 0.f, 0.f, 0.f, 0.f, 0.f, 0.f};
        const float* srow = &S[l16 * SSTR + kh * 1024];
        const float* vb   = V + (size_t)(kh * 1024) * DHEAD + d0 + l16;
        for (int kk = 0; kk < 1024; kk += 4) {
            v2f a = *(const v2f*)(srow + kk + 2 * lh);          // ds_load_b64 (SSTR even)
            const float* vrow = vb + (size_t)(kk + 2 * lh) * DHEAD;
            v2f b;
            b.x = vrow[0];
            b.y = vrow[DHEAD];
            acc = wmma_f32(a, b, acc);
        }
#pragma unroll
        for (int r = 0; r < 8; ++r)
            Pp[(kh * 16 + r + 8 * lh) * PSTR + d0 + l16] = acc[r];
    }
    __syncthreads();

    // ---- Phase 4: 2-way partial reduce + depthwise conv (from LDS) + store ----
    const int b_ = bh / HEADS;
    const int h  = bh % HEADS;
    const float* wres = resw + h * 33;
    for (int e = tid; e < 16 * DHEAD; e += 256) {
        const int m = e >> 6, d = e & 63;
        const float sum = Pp[m * PSTR + d] + Pp[(16 + m) * PSTR + d];
        float conv = 0.f;
#pragma unroll
        for (int t = 0; t < 33; ++t)
            conv += wres[t] * v_s[(m + t) * DHEAD + d];
        const int i = q0 + m;
        out[((size_t)(b_ * NSEQ + i)) * (HEADS * DHEAD) + h * DHEAD + d] = sum + conv;
    }
}

// =================================================================================
extern "C" void kernel_launch(void* const* d_in, const int* in_sizes, int n_in,
                              void* d_out, int out_size, void* d_ws, size_t ws_size,
                              hipStream_t stream) {
    (void)in_sizes; (void)n_in; (void)out_size; (void)ws_size;

    const float* x     = (const float*)d_in[0];
    const float* gamma = (const float*)d_in[1];
    const float* beta  = (const float*)d_in[2];
    const float* wqkv  = (const float*)d_in[3];
    const float* resw  = (const float*)d_in[4];
    float* out = (float*)d_out;

    // workspace: q | k | v, each [4][6][2048][64] fp32 (12.6 MB each)
    const size_t seg = (size_t)NBAT * HEADS * NSEQ * DHEAD;
    float* wsq = (float*)d_ws;
    float* wsk = wsq + seg;
    float* wsv = wsk + seg;

    // Kernel 1: LN + QKV
    dim3 g1(NQKV / 128, (NBAT * NSEQ) / 16);   // (9, 512)
    ps3_ln_qkv_kernel<<<g1, 256, 0, stream>>>(x, gamma, beta, wqkv, wsq, wsk, wsv);

    // Kernel 2: attention + conv. Dynamic LDS ~156 KB -> 2 WGs per 320 KB WGP.
    const size_t shmem =
        (size_t)(16 * SSTR + 16 * QSTR + 2 * 16 * PSTR + VWIN * DHEAD) * sizeof(float);
    (void)hipFuncSetAttribute((const void*)ps3_attn_kernel,
                              hipFuncAttributeMaxDynamicSharedMemorySize,
                              (int)shmem);   // idempotent, capture-safe
    dim3 g2(NSEQ / 16, NBAT * HEADS);          // (128, 24)
    ps3_attn_kernel<<<g2, 256, shmem, stream>>>(wsq, wsk, wsv, resw, out);
}